// DeformableConvBlock_16750372455125
// MI455X (gfx1250) — compile-verified
//
#include <hip/hip_runtime.h>

typedef __attribute__((ext_vector_type(2))) float v2f;
typedef __attribute__((ext_vector_type(8))) float v8f;

#define Bn 4
#define Cc 64
#define Hh 128
#define Ww 128
#define Oo 64
#define KKt 9
#define Kdim 576          // Cc * KKt
#define HW (Hh * Ww)      // 16384
#define TW 16             // pixels per wave tile
#define WAVES 4

// Workspace layout (floats):
//   Wt_off[9][32][32][2] : transposed, zero-padded offset-conv weights (18432 f)
//   Wt_def[9][32][64][2] : transposed deform weights                  (36864 f)
#define WTOFF_ELEMS (9 * 32 * 32 * 2)
#define WTDEF_ELEMS (9 * 32 * 64 * 2)

// ---------------- kernel 0: weight transpose to tap-major, paired-c ----------
__global__ __launch_bounds__(256) void transpose_weights_kernel(
    const float* __restrict__ w_off,
    const float* __restrict__ w_def,
    float* __restrict__ ws)
{
    const int i = blockIdx.x * 256 + threadIdx.x;
    if (i < WTOFF_ELEMS) {
        int e = i;
        const int half  = e & 1;  e >>= 1;
        const int row   = e & 31; e >>= 5;
        const int cpair = e & 31; e >>= 5;
        const int t     = e;                      // 0..8
        const int c     = cpair * 2 + half;
        ws[i] = (row < 2 * KKt) ? w_off[row * Kdim + c * KKt + t] : 0.0f;
    }
    if (i < WTDEF_ELEMS) {
        int e = i;
        const int half  = e & 1;  e >>= 1;
        const int row   = e & 63; e >>= 6;
        const int cpair = e & 31; e >>= 5;
        const int t     = e;
        const int c     = cpair * 2 + half;
        ws[WTOFF_ELEMS + i] = w_def[row * Kdim + c * KKt + t];
    }
}

// ---------------- kernel 1: fused deformable conv ---------------------------
__global__ __launch_bounds__(128) void deform_fused_kernel(
    const float* __restrict__ x,
    const float* __restrict__ b_off,
    const float* __restrict__ b_def,
    const float* __restrict__ wt_off,   // [9][32][32][2]
    const float* __restrict__ wt_def,   // [9][32][64][2]
    float* __restrict__ out)
{
    __shared__ float s_off[WAVES][2 * KKt][TW];        // 18 x 16 offsets per wave
    __shared__ float s_corner[WAVES][KKt * TW][8];     // 144 x (4 addr + 4 weight)

    const int tid  = threadIdx.x;
    const int lane = tid & 31;
    const int wv   = tid >> 5;

    // tile = 16 consecutive pixels of one row
    const int tile = blockIdx.x * WAVES + wv;   // 0..4095
    const int w0   = (tile & 7) * TW;
    const int h    = (tile >> 3) & (Hh - 1);
    const int b    = tile >> 10;

    const int col   = lane & 15;
    const int hi    = lane >> 4;      // which K-half this lane supplies
    const int khalf = hi * 2;
    const int wcol  = w0 + col;
    const float* __restrict__ xb = x + (size_t)b * Cc * HW;

    // ========== Stage A: offset conv (18 ch) via f32 WMMA, tap-major K ======
    v8f accA0 = {};
    v8f accA1 = {};
    for (int t = 0; t < KKt; ++t) {
        const int dy = t / 3 - 1;
        const int dx = t % 3 - 1;
        const int yy = h + dy;                 // wave-uniform
        const int xx = wcol + dx;              // per-lane
        const bool v = ((unsigned)yy < (unsigned)Hh) && ((unsigned)xx < (unsigned)Ww);
        const int addrBase = v ? (yy * Ww + xx) : 0;   // safe address when invalid
        const float* __restrict__ wtA = wt_off + (size_t)t * (32 * 32 * 2);
        for (int cp = 0; cp < 16; ++cp) {
            const int c0 = 4 * cp + khalf;
            float f0 = xb[(c0 << 14) + addrBase];
            float f1 = xb[((c0 + 1) << 14) + addrBase];
            f0 = v ? f0 : 0.0f;
            f1 = v ? f1 : 0.0f;
            v2f bb; bb.x = f0; bb.y = f1;
            const int cpair = 2 * cp + hi;
            v2f a0 = *(const v2f*)(wtA + ((size_t)cpair * 32 + col) * 2);
            v2f a1 = *(const v2f*)(wtA + ((size_t)cpair * 32 + 16 + col) * 2);
            accA0 = __builtin_amdgcn_wmma_f32_16x16x4_f32(false, a0, false, bb, (short)0, accA0, false, false);
            accA1 = __builtin_amdgcn_wmma_f32_16x16x4_f32(false, a1, false, bb, (short)0, accA1, false, false);
        }
    }
    // D layout: VGPR r -> row = r + 8*hi, col = lane%16
#pragma unroll
    for (int r = 0; r < 8; ++r) {
        const int o = r + 8 * hi;
        s_off[wv][o][col] = accA0[r] + b_off[o];
    }
    if (hi == 0) {
        s_off[wv][16][col] = accA1[0] + b_off[16];
        s_off[wv][17][col] = accA1[1] + b_off[17];
    }
    __syncthreads();

    // ===== Stage B: bilinear corner table (addr + masked weights) in LDS ====
    for (int e = lane; e < KKt * TW; e += 32) {
        const int k  = e >> 4;      // tap 0..8
        const int cc = e & 15;      // pixel in tile
        const float dyo = s_off[wv][2 * k + 0][cc];
        const float dxo = s_off[wv][2 * k + 1][cc];
        const float sy = (float)(h - 1 + (k / 3)) + dyo;
        const float sx = (float)(w0 + cc - 1 + (k % 3)) + dxo;
        const float y0f = floorf(sy), x0f = floorf(sx);
        const float wy = sy - y0f, wx = sx - x0f;
        const int y0 = (int)y0f, x0 = (int)x0f;
        const int yc0 = min(max(y0, 0), Hh - 1);
        const int yc1 = min(max(y0 + 1, 0), Hh - 1);
        const int xc0 = min(max(x0, 0), Ww - 1);
        const int xc1 = min(max(x0 + 1, 0), Ww - 1);
        const float vy0 = (y0 >= 0 && y0 <= Hh - 1) ? 1.0f : 0.0f;
        const float vy1 = (y0 + 1 >= 0 && y0 + 1 <= Hh - 1) ? 1.0f : 0.0f;
        const float vx0 = (x0 >= 0 && x0 <= Ww - 1) ? 1.0f : 0.0f;
        const float vx1 = (x0 + 1 >= 0 && x0 + 1 <= Ww - 1) ? 1.0f : 0.0f;
        float* ent = s_corner[wv][e];
        ((int*)ent)[0] = yc0 * Ww + xc0;
        ((int*)ent)[1] = yc0 * Ww + xc1;
        ((int*)ent)[2] = yc1 * Ww + xc0;
        ((int*)ent)[3] = yc1 * Ww + xc1;
        ent[4] = (1.0f - wy) * (1.0f - wx) * vy0 * vx0;
        ent[5] = (1.0f - wy) * wx * vy0 * vx1;
        ent[6] = wy * (1.0f - wx) * vy1 * vx0;
        ent[7] = wy * wx * vy1 * vx1;
    }
    __syncthreads();

    // ====== Stage C: main GEMM M=64, K=576 (tap-major), N=16, f32 WMMA ======
    v8f acc0 = {}, acc1 = {}, acc2 = {}, acc3 = {};
    for (int t = 0; t < KKt; ++t) {
        // corner entry is invariant for the whole tap
        const float* ent = s_corner[wv][t * TW + col];
        const int4   ai  = *(const int4*)ent;
        const float4 cw  = *(const float4*)(ent + 4);
        const float* __restrict__ wtC = wt_def + (size_t)t * (32 * 64 * 2);
        for (int cp = 0; cp < 16; ++cp) {
            const int c0 = 4 * cp + khalf;
            const float* xc0 = xb + (c0 << 14);
            const float* xc1 = xb + ((c0 + 1) << 14);
            v2f bb;
            bb.x = cw.x * xc0[ai.x] + cw.y * xc0[ai.y] + cw.z * xc0[ai.z] + cw.w * xc0[ai.w];
            bb.y = cw.x * xc1[ai.x] + cw.y * xc1[ai.y] + cw.z * xc1[ai.z] + cw.w * xc1[ai.w];
            const int cpair = 2 * cp + hi;
            const float* arow = wtC + (size_t)cpair * (64 * 2);
            v2f a0 = *(const v2f*)(arow + (col +  0) * 2);
            v2f a1 = *(const v2f*)(arow + (col + 16) * 2);
            v2f a2 = *(const v2f*)(arow + (col + 32) * 2);
            v2f a3 = *(const v2f*)(arow + (col + 48) * 2);
            acc0 = __builtin_amdgcn_wmma_f32_16x16x4_f32(false, a0, false, bb, (short)0, acc0, false, false);
            acc1 = __builtin_amdgcn_wmma_f32_16x16x4_f32(false, a1, false, bb, (short)0, acc1, false, false);
            acc2 = __builtin_amdgcn_wmma_f32_16x16x4_f32(false, a2, false, bb, (short)0, acc2, false, false);
            acc3 = __builtin_amdgcn_wmma_f32_16x16x4_f32(false, a3, false, bb, (short)0, acc3, false, false);
        }
    }

    // ---- bias + store: out[b, o, h, w0+col] ----
    float* ob = out + ((size_t)b * Oo) * HW + h * Ww + w0 + col;
#pragma unroll
    for (int r = 0; r < 8; ++r) {
        const int m0 = r + 8 * hi;
        ob[(size_t)(m0 +  0) * HW] = acc0[r] + b_def[m0 +  0];
        ob[(size_t)(m0 + 16) * HW] = acc1[r] + b_def[m0 + 16];
        ob[(size_t)(m0 + 32) * HW] = acc2[r] + b_def[m0 + 32];
        ob[(size_t)(m0 + 48) * HW] = acc3[r] + b_def[m0 + 48];
    }
}

extern "C" void kernel_launch(void* const* d_in, const int* in_sizes, int n_in,
                              void* d_out, int out_size, void* d_ws, size_t ws_size,
                              hipStream_t stream) {
    const float* x     = (const float*)d_in[0];
    const float* w_off = (const float*)d_in[1];
    const float* b_off = (const float*)d_in[2];
    const float* w_def = (const float*)d_in[3];
    const float* b_def = (const float*)d_in[4];
    float* out = (float*)d_out;
    float* ws  = (float*)d_ws;

    // kernel 0: transpose weights into workspace (tap-major, paired-c, coalesced)
    const int tr_threads = WTDEF_ELEMS;                 // 36864
    hipLaunchKernelGGL(transpose_weights_kernel,
                       dim3((tr_threads + 255) / 256), dim3(256), 0, stream,
                       w_off, w_def, ws);

    // kernel 1: fused offset-conv + bilinear sample + output GEMM
    const int tiles  = Bn * Hh * (Ww / TW);             // 4096 waves
    const int blocks = tiles / WAVES;                   // 1024 blocks of 128
    hipLaunchKernelGGL(deform_fused_kernel, dim3(blocks), dim3(128), 0, stream,
                       x, b_off, b_def, ws, ws + WTOFF_ELEMS, out);
}